// mLSTMLayer_18399639896251
// MI455X (gfx1250) — compile-verified
//
#include <hip/hip_runtime.h>
#include <math.h>

// ---------------------------------------------------------------------------
// mLSTM layer for gfx1250 (MI455X), fp32 end-to-end, chunkwise-parallel form.
//  K1: fused q/k/v/o projection GEMM            (V_WMMA_F32_16X16X4_F32)
//  K2: i/f gate projections (head-major output)
//  K3: gate scan: m (max,+ prefix) and per-chunk cumsum g, parallel prefix
//  K4: intra-chunk kernel, 1024 parallel blocks: QV^T, QK^T, (w.Sv)K, G, u
//  K5: inter-chunk kernel, 64 blocks x 16 sequential chunk steps: Q@C + update
//  K6: LayerNorm -> d_out
//  Tile staging uses GLOBAL_LOAD_ASYNC_TO_LDS_B128 (ASYNCcnt) where no
//  transpose is needed; LDS tiles are 16B-aligned per row for B128.
// ---------------------------------------------------------------------------

typedef __attribute__((ext_vector_type(2))) float v2f;
typedef __attribute__((ext_vector_type(8))) float v8f;

#define NTOK   4096          // B*T
#define INDIM  1024
#define HIDDIM 1024
#define NHEAD  16
#define HDIM   64
#define CHUNK  64
#define NCHUNK 16            // T / CHUNK
#define BHEADS 64            // B*H

#define TILE_M 64
#define TILE_N 128
#define KSTEP  32
#define LDA_P  36            // 144B rows: 16B aligned, bank-safe
#define LDB_P  36
#define LDC    68            // 272B rows: 16B aligned (async B128), bank-safe

__device__ __forceinline__ v8f wmma4(v2f a, v2f b, v8f c) {
    return __builtin_amdgcn_wmma_f32_16x16x4_f32(false, a, false, b, (short)0, c, false, false);
}

// Async 16B copy: global -> LDS, tracked by ASYNCcnt (per-lane addresses).
__device__ __forceinline__ void async_ld_b128(void* lds_ptr, const void* gaddr) {
    unsigned lds_off = (unsigned)(unsigned long long)lds_ptr;  // flat LDS addr[31:0]
    asm volatile("global_load_async_to_lds_b128 %0, %1, off"
                 :: "v"(lds_off), "v"(gaddr) : "memory");
}
__device__ __forceinline__ void wait_async0() {
    asm volatile("s_wait_asynccnt 0x0" ::: "memory");
}

// ---------------------------------------------------------------------------
// K1: fused projection GEMM.  grid = 64*32 = 2048 blocks, 256 threads.
// ---------------------------------------------------------------------------
__global__ __launch_bounds__(256)
void proj_gemm_kernel(const float* __restrict__ x,
                      const float* __restrict__ Wq, const float* __restrict__ Wk,
                      const float* __restrict__ Wv, const float* __restrict__ Wo,
                      const float* __restrict__ b_o,
                      float* __restrict__ outq, float* __restrict__ outk,
                      float* __restrict__ outv, float* __restrict__ outo)
{
    __shared__ float As[TILE_M][LDA_P];
    __shared__ float Bs[TILE_N][LDB_P];

    const int tid  = threadIdx.x;
    const int lane = tid & 31;
    const int wave = tid >> 5;

    const int bm = blockIdx.x >> 5;
    const int bn = blockIdx.x & 31;
    const int group    = bn >> 3;
    const int n_base_g = (bn & 7) * TILE_N;

    const float* W = (group == 0) ? Wq : (group == 1) ? Wk : (group == 2) ? Wv : Wo;
    float* outp    = (group == 0) ? outq : (group == 1) ? outk : (group == 2) ? outv : outo;

    const int wm = (wave >> 2) * 32;
    const int wn = (wave & 3)  * 32;
    const int half = lane >> 4;
    const int l16  = lane & 15;

    v8f acc[2][2] = {};

    const int ar = tid >> 2;
    const int ac = (tid & 3) * 8;
    const int bk = tid >> 3;
    const int bc = (tid & 7) * 16;

    for (int k0 = 0; k0 < INDIM; k0 += KSTEP) {
        // ---- stage A (64 x 32): async global -> LDS, no VGPR round trip ----
        {
            const float* gp = x + (size_t)(bm * TILE_M + ar) * INDIM + k0 + ac;
            async_ld_b128(&As[ar][ac],     gp);
            async_ld_b128(&As[ar][ac + 4], gp + 4);
        }
        // ---- stage B (32 x 128), transposed into LDS via VGPRs ----
        {
            const float* gp = W + (size_t)(k0 + bk) * HIDDIM + n_base_g + bc;
            if (k0 + KSTEP < INDIM)
                __builtin_prefetch(gp + (size_t)KSTEP * HIDDIM, 0, 0);
            float4 b0 = *(const float4*)(gp);
            float4 b1 = *(const float4*)(gp + 4);
            float4 b2 = *(const float4*)(gp + 8);
            float4 b3 = *(const float4*)(gp + 12);
            Bs[bc +  0][bk] = b0.x;  Bs[bc +  1][bk] = b0.y;
            Bs[bc +  2][bk] = b0.z;  Bs[bc +  3][bk] = b0.w;
            Bs[bc +  4][bk] = b1.x;  Bs[bc +  5][bk] = b1.y;
            Bs[bc +  6][bk] = b1.z;  Bs[bc +  7][bk] = b1.w;
            Bs[bc +  8][bk] = b2.x;  Bs[bc +  9][bk] = b2.y;
            Bs[bc + 10][bk] = b2.z;  Bs[bc + 11][bk] = b2.w;
            Bs[bc + 12][bk] = b3.x;  Bs[bc + 13][bk] = b3.y;
            Bs[bc + 14][bk] = b3.z;  Bs[bc + 15][bk] = b3.w;
        }
        wait_async0();
        __syncthreads();

        #pragma unroll
        for (int kk = 0; kk < KSTEP; kk += 4) {
            const int ak = kk + half * 2;
            v2f a0 = *(const v2f*)&As[wm +  0 + l16][ak];
            v2f a1 = *(const v2f*)&As[wm + 16 + l16][ak];
            v2f b0 = *(const v2f*)&Bs[wn +  0 + l16][ak];
            v2f b1 = *(const v2f*)&Bs[wn + 16 + l16][ak];
            acc[0][0] = wmma4(a0, b0, acc[0][0]);
            acc[0][1] = wmma4(a0, b1, acc[0][1]);
            acc[1][0] = wmma4(a1, b0, acc[1][0]);
            acc[1][1] = wmma4(a1, b1, acc[1][1]);
        }
        __syncthreads();
    }

    const int row0 = bm * TILE_M + wm;
    const int col0 = n_base_g + wn;
    #pragma unroll
    for (int i = 0; i < 2; ++i) {
        #pragma unroll
        for (int j = 0; j < 2; ++j) {
            const int col = col0 + j * 16 + l16;
            const float bo = (group == 3) ? b_o[col] : 0.0f;
            #pragma unroll
            for (int r = 0; r < 8; ++r) {
                const int row = row0 + i * 16 + r + half * 8;
                float val = acc[i][j][r];
                if (group == 0)      val *= 0.125f;                           // q / sqrt(D)
                else if (group == 3) val = 1.0f / (1.0f + expf(-(val + bo))); // sigmoid
                outp[(size_t)row * HIDDIM + col] = val;
            }
        }
    }
}

// ---------------------------------------------------------------------------
// K2: i/f projections, output HEAD-MAJOR: out[(b*16+h)*1024 + t].
// ---------------------------------------------------------------------------
__global__ __launch_bounds__(32)
void ifproj_kernel(const float* __restrict__ x,
                   const float* __restrict__ wi, const float* __restrict__ bi,
                   const float* __restrict__ wf, const float* __restrict__ bf,
                   float* __restrict__ outi, float* __restrict__ outf)
{
    const int row  = blockIdx.x;         // b*1024 + t
    const int b    = row >> 10;
    const int t    = row & 1023;
    const int lane = threadIdx.x;
    const int col  = lane & 15;
    const float* w = (lane < 16) ? wi : wf;
    const float* xr = x + (size_t)row * INDIM;
    float acc = 0.0f;
    #pragma unroll 4
    for (int kk = 0; kk < INDIM; ++kk)
        acc += xr[kk] * w[kk * NHEAD + col];
    acc += (lane < 16) ? bi[col] : bf[col];
    float* op = (lane < 16) ? outi : outf;
    op[(size_t)(b * NHEAD + col) * 1024 + t] = acc;
}

// ---------------------------------------------------------------------------
// K3: gate scan.  m_t = max(f_t + m_{t-1}, i_t) via (max,+) prefix:
//   g_j = per-chunk cumsum f;  m_j = g_j + max(m_entry, prefmax(i - g))
// ---------------------------------------------------------------------------
__global__ __launch_bounds__(64)
void gate_scan_kernel(const float* __restrict__ ihm, const float* __restrict__ fhm,
                      float* __restrict__ m_arr, float* __restrict__ g_arr)
{
    __shared__ float sarr[64];
    __shared__ float bc[2];
    const int bh  = blockIdx.x;
    const int tid = threadIdx.x;
    float me = 0.0f;
    for (int c = 0; c < NCHUNK; ++c) {
        const size_t ix = (size_t)bh * 1024 + c * CHUNK + tid;
        const float fv = fhm[ix];
        const float iv = ihm[ix];
        float g = fv;
        sarr[tid] = g; __syncthreads();
        #pragma unroll
        for (int off = 1; off < 64; off <<= 1) {
            const float other = (tid >= off) ? sarr[tid - off] : 0.0f;
            __syncthreads();
            g += other;
            sarr[tid] = g;
            __syncthreads();
        }
        if (tid == 63) bc[0] = g;
        float pm = iv - g;
        sarr[tid] = pm; __syncthreads();
        #pragma unroll
        for (int off = 1; off < 64; off <<= 1) {
            const float other = (tid >= off) ? sarr[tid - off] : -1e30f;
            __syncthreads();
            pm = fmaxf(pm, other);
            sarr[tid] = pm;
            __syncthreads();
        }
        if (tid == 63) bc[1] = pm;
        m_arr[ix] = g + fmaxf(me, pm);
        g_arr[ix] = g;
        __syncthreads();
        me = bc[0] + fmaxf(me, bc[1]);
        __syncthreads();
    }
}

// ---------------------------------------------------------------------------
// K4: intra-chunk kernel.  grid = 1024 blocks, 256 threads.
// ---------------------------------------------------------------------------
__global__ __launch_bounds__(256)
void chunk_intra_kernel(const float* __restrict__ qb, const float* __restrict__ kb,
                        const float* __restrict__ vb, const float* __restrict__ ihm,
                        const float* __restrict__ g_arr, const float* __restrict__ m_arr,
                        float* __restrict__ Pbuf, float* __restrict__ denbuf,
                        float* __restrict__ Gbuf, float* __restrict__ ubuf)
{
    __shared__ float Qs[64][LDC];   // later reused as W (weighted Sv)
    __shared__ float Ks[64][LDC];   // later scaled by wg
    __shared__ float Vs[64][LDC];
    __shared__ float sg[64], sm[64], sii[64], wgl[64];
    __shared__ float denp[2][64];
    __shared__ float red[256];

    const int tid  = threadIdx.x;
    const int lane = tid & 31;
    const int wave = tid >> 5;
    const int l16  = lane & 15;
    const int half = lane >> 4;

    const int bh = blockIdx.x >> 4;
    const int c  = blockIdx.x & 15;
    const int b  = bh >> 4, h = bh & 15;

    // ---- stage Q,K,V chunk tiles: async global -> LDS ----
    {
        const int r  = tid >> 2;
        const int c0 = (tid & 3) * 16;
        const size_t gbase = ((size_t)(b * 1024 + c * CHUNK + r)) * HIDDIM + h * HDIM + c0;
        #pragma unroll
        for (int z = 0; z < 4; ++z) {
            async_ld_b128(&Qs[r][c0 + z * 4], qb + gbase + z * 4);
            async_ld_b128(&Ks[r][c0 + z * 4], kb + gbase + z * 4);
            async_ld_b128(&Vs[r][c0 + z * 4], vb + gbase + z * 4);
        }
    }
    if (tid < 64) {
        const size_t sb = (size_t)bh * 1024 + c * CHUNK + tid;
        sg[tid]  = g_arr[sb];
        sm[tid]  = m_arr[sb];
        sii[tid] = ihm[sb];
    }
    wait_async0();
    __syncthreads();
    if (tid < 64)
        wgl[tid] = expf(sg[63] - sg[tid] + sii[tid] - sm[63]);   // state-inc weights

    const int tm  = wave >> 1;
    const int m0  = tm * 16;
    const int tn0 = (wave & 1) * 32;

    // ---- GEMM 1+2 fused: Sv = Q V^T, Sk = Q K^T ----
    v8f sv0 = {}, sv1 = {}, sk0 = {}, sk1 = {};
    #pragma unroll
    for (int kk = 0; kk < 64; kk += 4) {
        const int ak = kk + half * 2;
        v2f a   = *(const v2f*)&Qs[m0 + l16][ak];
        v2f bv0 = *(const v2f*)&Vs[tn0 + l16][ak];
        v2f bv1 = *(const v2f*)&Vs[tn0 + 16 + l16][ak];
        v2f bk0 = *(const v2f*)&Ks[tn0 + l16][ak];
        v2f bk1 = *(const v2f*)&Ks[tn0 + 16 + l16][ak];
        sv0 = wmma4(a, bv0, sv0);  sv1 = wmma4(a, bv1, sv1);
        sk0 = wmma4(a, bk0, sk0);  sk1 = wmma4(a, bk1, sk1);
    }
    __syncthreads();   // all Qs reads complete before overwrite

    // ---- causal decay weights, W into Qs, den_intra reduce ----
    #pragma unroll
    for (int r = 0; r < 8; ++r) {
        const int j  = m0 + r + half * 8;
        const int i0 = tn0 + l16;
        const int i1 = i0 + 16;
        const float gj = sg[j], mj = sm[j];
        const float w0 = (i0 <= j) ? expf(gj - sg[i0] + sii[i0] - mj) : 0.0f;
        const float w1 = (i1 <= j) ? expf(gj - sg[i1] + sii[i1] - mj) : 0.0f;
        Qs[j][i0] = w0 * sv0[r];
        Qs[j][i1] = w1 * sv1[r];
        float dp = w0 * sk0[r] + w1 * sk1[r];
        #pragma unroll
        for (int off = 8; off; off >>= 1) dp += __shfl_xor(dp, off, 32);
        if (l16 == 0) denp[wave & 1][j] = dp;
    }
    __syncthreads();
    if (tid < 64)
        denbuf[(size_t)bh * 1024 + c * CHUNK + tid] = denp[0][tid] + denp[1][tid];

    // ---- GEMM 3: P = W @ K  (B from row-major Ks via two scalar loads) ----
    v8f p0 = {}, p1 = {};
    #pragma unroll
    for (int kk = 0; kk < 64; kk += 4) {
        const int ak = kk + half * 2;
        v2f a = *(const v2f*)&Qs[m0 + l16][ak];
        v2f b0, b1;
        b0.x = Ks[ak][tn0 + l16];       b0.y = Ks[ak + 1][tn0 + l16];
        b1.x = Ks[ak][tn0 + 16 + l16];  b1.y = Ks[ak + 1][tn0 + 16 + l16];
        p0 = wmma4(a, b0, p0);  p1 = wmma4(a, b1, p1);
    }
    {
        const size_t pb = (size_t)blockIdx.x * 4096;   // (bh*16+c) * 64*64
        #pragma unroll
        for (int r = 0; r < 8; ++r) {
            const int j = m0 + r + half * 8;
            Pbuf[pb + j * 64 + tn0 + l16]      = p0[r];
            Pbuf[pb + j * 64 + tn0 + 16 + l16] = p1[r];
        }
    }
    __syncthreads();

    // ---- scale K rows by wg, column sums -> u ----
    #pragma unroll
    for (int z = 0; z < 16; ++z) {
        const int flat = tid + z * 256;
        Ks[flat >> 6][flat & 63] *= wgl[flat >> 6];
    }
    __syncthreads();
    {
        const int d = tid & 63, grp = tid >> 6;
        float s = 0.0f;
        #pragma unroll
        for (int z = 0; z < 16; ++z) s += Ks[grp * 16 + z][d];
        red[tid] = s;
    }
    __syncthreads();
    if (tid < 64)
        ubuf[(size_t)blockIdx.x * 64 + tid] = red[tid] + red[64 + tid] + red[128 + tid] + red[192 + tid];

    // ---- GEMM 4: G^T[e,d] = sum_i wg_i K[i,e] V[i,d] ----
    v8f gg0 = {}, gg1 = {};
    #pragma unroll
    for (int kk = 0; kk < 64; kk += 4) {
        const int ak = kk + half * 2;
        v2f a, b0, b1;
        a.x  = Ks[ak][m0 + l16];        a.y  = Ks[ak + 1][m0 + l16];        // A[e][i]
        b0.x = Vs[ak][tn0 + l16];       b0.y = Vs[ak + 1][tn0 + l16];       // B[i][d]
        b1.x = Vs[ak][tn0 + 16 + l16];  b1.y = Vs[ak + 1][tn0 + 16 + l16];
        gg0 = wmma4(a, b0, gg0);  gg1 = wmma4(a, b1, gg1);
    }
    {
        const size_t gb = (size_t)blockIdx.x * 4096;
        #pragma unroll
        for (int r = 0; r < 8; ++r) {
            const int e = m0 + r + half * 8;
            Gbuf[gb + e * 64 + tn0 + l16]      = gg0[r];
            Gbuf[gb + e * 64 + tn0 + 16 + l16] = gg1[r];
        }
    }
}

// ---------------------------------------------------------------------------
// K5: inter-chunk kernel.  grid = 64 (b,h), 256 threads, 16 sequential chunks.
// ---------------------------------------------------------------------------
__global__ __launch_bounds__(256)
void chunk_inter_kernel(const float* __restrict__ qb, const float* __restrict__ ob,
                        const float* __restrict__ g_arr, const float* __restrict__ m_arr,
                        const float* __restrict__ Gbuf, const float* __restrict__ ubuf,
                        const float* __restrict__ denbuf,
                        float* __restrict__ Hbuf)   // in: P_intra, out: h
{
    __shared__ float Ct[64][LDC];
    __shared__ float Qs[64][LDC];
    __shared__ float nst[64];
    __shared__ float sg[64], sm[64], dl[64];
    __shared__ float red[256];

    const int tid  = threadIdx.x;
    const int lane = tid & 31, wave = tid >> 5;
    const int l16  = lane & 15, half = lane >> 4;
    const int bh = blockIdx.x;
    const int b  = bh >> 4, h = bh & 15;
    const int tm = wave >> 1, m0 = tm * 16, tn0 = (wave & 1) * 32;

    #pragma unroll
    for (int z = 0; z < 16; ++z) {
        const int flat = tid + z * 256;
        Ct[flat >> 6][flat & 63] = 0.0f;
    }
    if (tid < 64) nst[tid] = 0.0f;

    for (int c = 0; c < NCHUNK; ++c) {
        __syncthreads();
        {
            const int r = tid >> 2, c0 = (tid & 3) * 16;
            const size_t gb2 = ((size_t)(b * 1024 + c * CHUNK + r)) * HIDDIM + h * HDIM + c0;
            #pragma unroll
            for (int z = 0; z < 4; ++z)
                async_ld_b128(&Qs[r][c0 + z * 4], qb + gb2 + z * 4);
        }
        if (tid < 64) {
            const size_t sb = (size_t)bh * 1024 + c * CHUNK + tid;
            sg[tid] = g_arr[sb];
            sm[tid] = m_arr[sb];
        }
        wait_async0();
        __syncthreads();
        const float Mc = (c == 0) ? 0.0f : m_arr[(size_t)bh * 1024 + c * CHUNK - 1];

        // den-inter partials: n . q_j
        {
            const int j = tid & 63, grp = tid >> 6;
            float s = 0.0f;
            #pragma unroll
            for (int z = 0; z < 16; ++z) s += nst[grp * 16 + z] * Qs[j][grp * 16 + z];
            red[tid] = s;
        }
        // inter[j,e] = sum_d Q[j,d] * Ctilde[d,e]   (Ct holds [e][d])
        v8f a0 = {}, a1 = {};
        #pragma unroll
        for (int kk = 0; kk < 64; kk += 4) {
            const int ak = kk + half * 2;
            v2f av = *(const v2f*)&Qs[m0 + l16][ak];
            v2f b0 = *(const v2f*)&Ct[tn0 + l16][ak];
            v2f b1 = *(const v2f*)&Ct[tn0 + 16 + l16][ak];
            a0 = wmma4(av, b0, a0);  a1 = wmma4(av, b1, a1);
        }
        __syncthreads();
        if (tid < 64) dl[tid] = red[tid] + red[64 + tid] + red[128 + tid] + red[192 + tid];

        // state update: Ctilde = decay*Ctilde + G_c ; n = decay*n + u_c
        const float decay = expf(sg[63] + Mc - sm[63]);
        {
            const size_t gbase = (size_t)(bh * NCHUNK + c) * 4096;
            #pragma unroll
            for (int z = 0; z < 16; ++z) {
                const int flat = tid + z * 256;
                Ct[flat >> 6][flat & 63] = decay * Ct[flat >> 6][flat & 63] + Gbuf[gbase + flat];
            }
            if (tid < 64)
                nst[tid] = decay * nst[tid] + ubuf[(size_t)(bh * NCHUNK + c) * 64 + tid];
        }
        __syncthreads();

        // combine: h = o .* (scl*inter + P_intra) / max(|scl*deninter + den_intra|, 1)
        {
            const size_t pb = (size_t)(bh * NCHUNK + c) * 4096;
            #pragma unroll
            for (int r = 0; r < 8; ++r) {
                const int j = m0 + r + half * 8;
                const float scl = expf(sg[j] + Mc - sm[j]);
                float den = scl * dl[j] + denbuf[(size_t)bh * 1024 + c * CHUNK + j];
                den = fmaxf(fabsf(den), 1.0f);
                const size_t orow = ((size_t)(b * 1024 + c * CHUNK + j)) * HIDDIM + h * HDIM;
                const int e0 = tn0 + l16, e1 = e0 + 16;
                const float n0 = scl * a0[r] + Hbuf[pb + j * 64 + e0];
                const float n1 = scl * a1[r] + Hbuf[pb + j * 64 + e1];
                Hbuf[pb + j * 64 + e0] = ob[orow + e0] * n0 / den;
                Hbuf[pb + j * 64 + e1] = ob[orow + e1] * n1 / den;
            }
        }
    }
}

// ---------------------------------------------------------------------------
// K6: LayerNorm over head-major h.  grid = 4096 token rows, 256 threads.
// ---------------------------------------------------------------------------
__global__ __launch_bounds__(256)
void ln_kernel(const float* __restrict__ hb, const float* __restrict__ gamma,
               const float* __restrict__ beta, float* __restrict__ out)
{
    __shared__ float rs[8], rs2[8];
    const int row = blockIdx.x;          // b*1024 + t
    const int b = row >> 10, t = row & 1023;
    const int tid = threadIdx.x;

    float vals[4];
    float s = 0.0f, s2 = 0.0f;
    #pragma unroll
    for (int i = 0; i < 4; ++i) {
        const int cH = tid + i * 256, h = cH >> 6, e = cH & 63;
        const float vv = hb[((size_t)(b * NHEAD + h) * 1024 + t) * HDIM + e];
        vals[i] = vv;
        s += vv;  s2 += vv * vv;
    }
    #pragma unroll
    for (int off = 16; off; off >>= 1) {
        s  += __shfl_xor(s,  off, 32);
        s2 += __shfl_xor(s2, off, 32);
    }
    const int wave = tid >> 5, lane = tid & 31;
    if (lane == 0) { rs[wave] = s; rs2[wave] = s2; }
    __syncthreads();
    float ts = 0.0f, ts2 = 0.0f;
    #pragma unroll
    for (int w = 0; w < 8; ++w) { ts += rs[w]; ts2 += rs2[w]; }
    const float mu  = ts * (1.0f / 1024.0f);
    const float var = ts2 * (1.0f / 1024.0f) - mu * mu;
    const float inv = rsqrtf(var + 1e-5f);
    #pragma unroll
    for (int i = 0; i < 4; ++i) {
        const int cH = tid + i * 256;
        out[(size_t)row * HIDDIM + cH] = (vals[i] - mu) * inv * gamma[cH] + beta[cH];
    }
}

// ---------------------------------------------------------------------------
extern "C" void kernel_launch(void* const* d_in, const int* in_sizes, int n_in,
                              void* d_out, int out_size, void* d_ws, size_t ws_size,
                              hipStream_t stream)
{
    const float* x     = (const float*)d_in[0];
    const float* Wq    = (const float*)d_in[1];
    const float* Wk    = (const float*)d_in[2];
    const float* Wv    = (const float*)d_in[3];
    const float* Wo    = (const float*)d_in[4];
    const float* b_o   = (const float*)d_in[5];
    const float* w_i   = (const float*)d_in[6];
    const float* b_i   = (const float*)d_in[7];
    const float* w_f   = (const float*)d_in[8];
    const float* b_f   = (const float*)d_in[9];
    const float* gamma = (const float*)d_in[10];
    const float* beta  = (const float*)d_in[11];

    // workspace layout (floats); total ~= 24.4M floats ~= 98 MB
    float* ws = (float*)d_ws;
    const size_t big = (size_t)NTOK * HIDDIM;   // 4M
    const size_t sml = (size_t)BHEADS * 1024;   // 64K
    float* q_buf = ws;
    float* k_buf = q_buf + big;
    float* v_buf = k_buf + big;
    float* o_buf = v_buf + big;
    float* i_buf = o_buf + big;      // head-major (bh, t)
    float* f_buf = i_buf + sml;
    float* m_buf = f_buf + sml;
    float* g_buf = m_buf + sml;
    float* P_buf = g_buf + sml;      // (bh, t, e) — becomes h after K5
    float* d_buf = P_buf + big;      // den_intra (bh, t)
    float* G_buf = d_buf + sml;      // (bh*16+c, e, d)
    float* u_buf = G_buf + big;      // (bh*16+c, d)

    proj_gemm_kernel<<<dim3(2048), dim3(256), 0, stream>>>(
        x, Wq, Wk, Wv, Wo, b_o, q_buf, k_buf, v_buf, o_buf);

    ifproj_kernel<<<dim3(NTOK), dim3(32), 0, stream>>>(
        x, w_i, b_i, w_f, b_f, i_buf, f_buf);

    gate_scan_kernel<<<dim3(BHEADS), dim3(64), 0, stream>>>(
        i_buf, f_buf, m_buf, g_buf);

    chunk_intra_kernel<<<dim3(BHEADS * NCHUNK), dim3(256), 0, stream>>>(
        q_buf, k_buf, v_buf, i_buf, g_buf, m_buf, P_buf, d_buf, G_buf, u_buf);

    chunk_inter_kernel<<<dim3(BHEADS), dim3(256), 0, stream>>>(
        q_buf, o_buf, g_buf, m_buf, G_buf, u_buf, d_buf, P_buf);

    ln_kernel<<<dim3(NTOK), dim3(256), 0, stream>>>(
        P_buf, gamma, beta, (float*)d_out);
}